// DirGCNConv_55972013801706
// MI455X (gfx1250) — compile-verified
//
#include <hip/hip_runtime.h>

// Problem constants (match reference)
#define N_NODES 100000
#define N_EDGES 800000
#define CH      128
#define ALPHA_F 0.5f

typedef float v2f __attribute__((ext_vector_type(2)));
typedef float v8f __attribute__((ext_vector_type(8)));

__device__ __forceinline__ void atomAddF(float* p, float v) {
    __hip_atomic_fetch_add(p, v, __ATOMIC_RELAXED, __HIP_MEMORY_SCOPE_AGENT);
}

// ---------------- utility kernels ----------------

__global__ void k_zero_f32(float* __restrict__ p, int n) {
    int i = blockIdx.x * blockDim.x + threadIdx.x;
    if (i < n) p[i] = 0.0f;
}

// Pre-swizzle W (K-major 128x128) into the exact B-fragment layout for
// V_WMMA_F32_16X16X4_F32:
//   B VGPR0: lanes 0-15 -> K=k+0, N=n0+lane ; lanes 16-31 -> K=k+2
//   B VGPR1: lanes 0-15 -> K=k+1            ; lanes 16-31 -> K=k+3
// Packed so the inner loop does one contiguous 8-byte load per lane.
__global__ void k_pack_w(const float* __restrict__ Wsd, const float* __restrict__ Wds,
                         float* __restrict__ Psd, float* __restrict__ Pds) {
    int i = blockIdx.x * blockDim.x + threadIdx.x;   // over 128*128 elements
    if (i >= CH * CH) return;
    int kk = i >> 7;            // K index (row of W)
    int n  = i & 127;           // N index (col of W)
    int kstep = kk >> 2;        // which WMMA (K/4)
    int hi    = (kk >> 1) & 1;  // lane-half
    int odd   = kk & 1;         // which of the two floats in the lane
    int lane  = hi * 16 + (n & 15);
    int group = n >> 4;         // which 16-col wave stripe
    int dst   = (group * 32 + kstep) * 64 + lane * 2 + odd;
    Psd[dst] = Wsd[i];
    Pds[dst] = Wds[i];
}

__global__ void k_degrees(const int* __restrict__ row, const int* __restrict__ col,
                          float* __restrict__ odeg, float* __restrict__ ideg) {
    int e = blockIdx.x * blockDim.x + threadIdx.x;
    if (e < N_EDGES) {
        atomAddF(&odeg[row[e]], 1.0f);
        atomAddF(&ideg[col[e]], 1.0f);
    }
}

__global__ void k_inv_sqrt(float* __restrict__ p, int n) {
    int i = blockIdx.x * blockDim.x + threadIdx.x;
    if (i < n) {
        float v = p[i];
        p[i] = (v > 0.0f) ? rsqrtf(v) : 0.0f;
    }
}

__global__ void k_init_out(float* __restrict__ out,
                           const float* __restrict__ bsd, const float* __restrict__ bds) {
    int i = blockIdx.x * blockDim.x + threadIdx.x;
    if (i < N_NODES * CH) {
        int c = i & 127;
        out[i] = ALPHA_F * bds[c] + (1.0f - ALPHA_F) * bsd[c];
    }
}

// ---------------- FP32 WMMA GEMM: h = x @ W ----------------
// Block: 256 threads = 8 waves. Each block computes a 16x128 tile of h for one
// of the two weight matrices (blockIdx.y). Each wave owns a 16x16 sub-tile and
// runs 32 x V_WMMA_F32_16X16X4_F32 over K=128.
__global__ __launch_bounds__(256) void k_gemm_wmma(
    const float* __restrict__ x,
    const float* __restrict__ Psd, const float* __restrict__ Pds,
    float* __restrict__ hsd, float* __restrict__ hds) {

    // 16 rows x 128 cols of x, skewed stride 132 (4-word skew => the A-fragment
    // b64 reads hit 64 distinct banks across the wave: conflict-free).
    __shared__ float lx[16 * 132];

    const int row0 = blockIdx.x * 16;

    {   // cooperative tile load: each thread copies 8 consecutive floats
        int t  = threadIdx.x;
        int r  = t >> 4;
        int cb = (t & 15) * 8;
        const float* src = x + (size_t)(row0 + r) * CH + cb;
        float* dst = &lx[r * 132 + cb];
        #pragma unroll
        for (int j = 0; j < 8; ++j) dst[j] = src[j];
    }
    __syncthreads();

    const int wave = threadIdx.x >> 5;
    const int lane = threadIdx.x & 31;
    const int m  = lane & 15;
    const int hi = lane >> 4;

    const float* P = blockIdx.y ? Pds : Psd;
    float*       h = blockIdx.y ? hds : hsd;

    // A fragment source: lane reads A[M=m][K=4*ks+2*hi .. +1] (contiguous pair)
    const float* arow = &lx[m * 132 + 2 * hi];
    // B fragment source: pre-swizzled, one v2f per lane per K-step
    const float* brow = P + (size_t)(wave * 32) * 64 + lane * 2;

    v8f acc = {};
    #pragma unroll
    for (int ks = 0; ks < 32; ++ks) {
        v2f a = *(const v2f*)(arow + ks * 4);
        v2f b = *(const v2f*)(brow + ks * 64);
        // D = A(16x4,f32) * B(4x16,f32) + C(16x16,f32)  -- exact FP32
        acc = __builtin_amdgcn_wmma_f32_16x16x4_f32(
            /*neg_a=*/false, a, /*neg_b=*/false, b,
            /*c_mod=*/(short)0, acc, /*reuse_a=*/false, /*reuse_b=*/false);
    }

    // C/D layout: lanes 0-15 -> M=j, N=m ; lanes 16-31 -> M=8+j, N=m
    float* hp = h + (size_t)row0 * CH + wave * 16 + m;
    #pragma unroll
    for (int j = 0; j < 8; ++j)
        hp[(size_t)(j + 8 * hi) * CH] = acc[j];
}

// ---------------- fused bidirectional edge scatter ----------------
// One wave per edge; lane handles 4 channels via float4.
// Uses w_t == w (norms for the transposed graph algebraically equal forward).
__global__ __launch_bounds__(256) void k_edge_scatter(
    const int* __restrict__ row, const int* __restrict__ col,
    const float* __restrict__ hsd, const float* __restrict__ hds,
    const float* __restrict__ oinv, const float* __restrict__ iinv,
    float* __restrict__ out) {

    int e    = (blockIdx.x * blockDim.x + threadIdx.x) >> 5;
    int lane = threadIdx.x & 31;
    if (e >= N_EDGES) return;

    int r = row[e];
    int c = col[e];
    float w  = oinv[r] * iinv[c];
    float wa = ALPHA_F * w;          // dst->src direction (x_in), lands on row
    float wb = (1.0f - ALPHA_F) * w; // src->dst direction (x_out), lands on col

    const float4 hd = *(const float4*)(hds + (size_t)c * CH + lane * 4);
    const float4 hs = *(const float4*)(hsd + (size_t)r * CH + lane * 4);

    float* outr = out + (size_t)r * CH + lane * 4;
    float* outc = out + (size_t)c * CH + lane * 4;

    atomAddF(outr + 0, wa * hd.x);
    atomAddF(outr + 1, wa * hd.y);
    atomAddF(outr + 2, wa * hd.z);
    atomAddF(outr + 3, wa * hd.w);

    atomAddF(outc + 0, wb * hs.x);
    atomAddF(outc + 1, wb * hs.y);
    atomAddF(outc + 2, wb * hs.z);
    atomAddF(outc + 3, wb * hs.w);
}

// ---------------- launch ----------------

extern "C" void kernel_launch(void* const* d_in, const int* in_sizes, int n_in,
                              void* d_out, int out_size, void* d_ws, size_t ws_size,
                              hipStream_t stream) {
    const float* x   = (const float*)d_in[0];
    const int*   ei  = (const int*)  d_in[1];   // (2, E)
    const float* Wsd = (const float*)d_in[2];
    const float* bsd = (const float*)d_in[3];
    const float* Wds = (const float*)d_in[4];
    const float* bds = (const float*)d_in[5];
    float* out = (float*)d_out;

    const int* row = ei;            // edge_index[0]
    const int* col = ei + N_EDGES;  // edge_index[1]

    // Workspace layout (floats):
    //   hsd [N*CH] | hds [N*CH] | oinv [N] | iinv [N] | Psd [CH*CH] | Pds [CH*CH]
    float* hsd  = (float*)d_ws;
    float* hds  = hsd  + (size_t)N_NODES * CH;
    float* oinv = hds  + (size_t)N_NODES * CH;
    float* iinv = oinv + N_NODES;
    float* Psd  = iinv + N_NODES;
    float* Pds  = Psd  + CH * CH;
    (void)ws_size; (void)in_sizes; (void)n_in; (void)out_size;

    // 1) zero degree accumulators (oinv/iinv are contiguous: 2N floats)
    k_zero_f32<<<(2 * N_NODES + 255) / 256, 256, 0, stream>>>(oinv, 2 * N_NODES);

    // 2) pre-swizzle both weight matrices into B-fragment layout
    k_pack_w<<<(CH * CH + 255) / 256, 256, 0, stream>>>(Wsd, Wds, Psd, Pds);

    // 3) degree counts via f32 atomics
    k_degrees<<<(N_EDGES + 255) / 256, 256, 0, stream>>>(row, col, oinv, iinv);

    // 4) in-place deg -> rsqrt(deg) (0 where deg==0)
    k_inv_sqrt<<<(2 * N_NODES + 255) / 256, 256, 0, stream>>>(oinv, 2 * N_NODES);

    // 5) both GEMMs: h_sd = x@W_sd, h_ds = x@W_ds (blockIdx.y selects matrix)
    dim3 ggrid(N_NODES / 16, 2);   // 100000 % 16 == 0
    k_gemm_wmma<<<ggrid, 256, 0, stream>>>(x, Psd, Pds, hsd, hds);

    // 6) initialize output with blended biases
    k_init_out<<<(N_NODES * CH + 255) / 256, 256, 0, stream>>>(out, bsd, bds);

    // 7) fused bidirectional normalized scatter-add (one wave per edge)
    long long sthreads = (long long)N_EDGES * 32;
    k_edge_scatter<<<(int)((sthreads + 255) / 256), 256, 0, stream>>>(
        row, col, hsd, hds, oinv, iinv, out);
}